// TopKNonlinearMixGate_8091718385705
// MI455X (gfx1250) — compile-verified
//
#include <hip/hip_runtime.h>
#include <cstddef>

typedef __attribute__((ext_vector_type(2))) float v2f;
typedef __attribute__((ext_vector_type(8))) float v8f;

#define NB 32
#define NC 64
#define NH 64
#define NW 64
#define NE 8
#define HW 4096   // NH*NW
#define NPOS 3844.0f  // 62*62 valid positions

// ---------------------------------------------------------------------------
// Kernel 1: per (b,c) compute the nine 62x62 window sums S[b,c,i,j] from the
// full-plane sum, border row/col sums, and 4x4 corner elements.
// One block per (b,c); 64 threads, one row each.
// ---------------------------------------------------------------------------
__global__ void k_window_sums(const float* __restrict__ x, float* __restrict__ S)
{
    const int bc  = blockIdx.x;        // 0 .. NB*NC-1
    const int tid = threadIdx.x;       // 0 .. 63 == row h
    const float* xp = x + (size_t)bc * HW + (size_t)tid * NW;

    __shared__ float rowsum[64];
    __shared__ float edge[64][4];      // cols {0, 1, 62, 63}
    __shared__ float csum[4];
    __shared__ float Tsum;

    float s = 0.0f;
    #pragma unroll
    for (int w = 0; w < NW; w += 4) {
        float4 v = *(const float4*)(xp + w);
        s += v.x + v.y + v.z + v.w;
    }
    rowsum[tid]  = s;
    edge[tid][0] = xp[0];
    edge[tid][1] = xp[1];
    edge[tid][2] = xp[62];
    edge[tid][3] = xp[63];
    __syncthreads();

    if (tid < 4) {
        float cs = 0.0f;
        for (int h = 0; h < 64; ++h) cs += edge[h][tid];
        csum[tid] = cs;
    }
    if (tid == 4) {
        float t = 0.0f;
        for (int h = 0; h < 64; ++h) t += rowsum[h];
        Tsum = t;
    }
    __syncthreads();

    if (tid < 9) {
        const int i = tid / 3, j = tid % 3;
        // excluded rows: i==0 -> {62,63}, i==1 -> {0,63}, i==2 -> {0,1}
        const int ra = (i == 0) ? 62 : 0;
        const int rb = (i == 2) ? 1  : 63;
        // excluded cols map to edge slots {0:c0, 1:c1, 2:c62, 3:c63}
        const int ca = (j == 0) ? 2 : 0;
        const int cb = (j == 2) ? 1 : 3;
        float sv = Tsum - rowsum[ra] - rowsum[rb] - csum[ca] - csum[cb]
                 + edge[ra][ca] + edge[ra][cb] + edge[rb][ca] + edge[rb][cb];
        S[(size_t)bc * 9 + tid] = sv;
    }
}

// ---------------------------------------------------------------------------
// Kernel 2: per-batch gating code, softmax, top-1. One block per b, 64 threads
// (one input channel each). Writes scale[b], eidx[b] to ws and the dense
// expert_weights [B,E] tail of d_out.
// ---------------------------------------------------------------------------
__global__ void k_gate_select(const float* __restrict__ S,
                              const float* __restrict__ gw,   // [E,C,3,3]
                              const float* __restrict__ gb,   // [E]
                              float* __restrict__ ew_out,     // [B,E] tail of d_out
                              float* __restrict__ scale,      // [B]
                              int*   __restrict__ eidx)       // [B]
{
    const int b = blockIdx.x;
    const int c = threadIdx.x;   // 0..63

    float sv[9];
    const float* sp = S + ((size_t)b * NC + c) * 9;
    #pragma unroll
    for (int t = 0; t < 9; ++t) sv[t] = sp[t];

    __shared__ float red[NE][64];
    #pragma unroll
    for (int e = 0; e < NE; ++e) {
        const float* wp = gw + ((size_t)e * NC + c) * 9;
        float p = 0.0f;
        #pragma unroll
        for (int t = 0; t < 9; ++t) p += wp[t] * sv[t];
        red[e][c] = p;
    }
    __syncthreads();

    __shared__ float gc[NE];
    if (c < NE) {
        float a = gb[c] * NPOS;
        for (int k = 0; k < 64; ++k) a += red[c][k];
        gc[c] = a;
    }
    __syncthreads();

    __shared__ float s_top;
    __shared__ int   s_idx;
    if (c == 0) {
        float m = gc[0]; int mi = 0;
        #pragma unroll
        for (int e = 1; e < NE; ++e) if (gc[e] > m) { m = gc[e]; mi = e; }
        float sum = 0.0f;
        #pragma unroll
        for (int e = 0; e < NE; ++e) sum += __expf(gc[e] - m);
        float top = 1.0f / sum;     // softmax value at the argmax
        s_top = top; s_idx = mi;
        scale[b] = top;
        eidx[b]  = mi;
    }
    __syncthreads();
    if (c < NE) ew_out[(size_t)b * NE + c] = (c == s_idx) ? s_top : 0.0f;
}

// ---------------------------------------------------------------------------
// Kernel 3: out[b] = scale[b] * W[e_b] (64x64) @ X[b] (64x4096) using
// V_WMMA_F32_16X16X4_F32. One wave per 16x16 output tile; 8 waves/block.
// Tiles: NB * (64/16) * (4096/16) = 32768 -> 4096 blocks of 256 threads.
// ---------------------------------------------------------------------------
__global__ void k_expert_gemm(const float* __restrict__ x,      // [B,C,HW]
                              const float* __restrict__ Wx,     // [E,C,C] (out,in)
                              const float* __restrict__ scale,  // [B]
                              const int*   __restrict__ eidx,   // [B]
                              float* __restrict__ out)          // [B,C,HW]
{
    const int wave = threadIdx.x >> 5;
    const int lane = threadIdx.x & 31;
    const int tile = blockIdx.x * 8 + wave;     // 0 .. 32767
    const int b    = tile >> 10;                // 1024 tiles per batch
    const int t    = tile & 1023;
    const int m0   = (t >> 8) << 4;             // output-channel tile: 0,16,32,48
    const int n0   = (t & 255) << 4;            // spatial tile

    const int e  = eidx[b];
    const float sc = scale[b];

    const int half = lane >> 4;                 // K half-select per ISA layout
    const int mr   = lane & 15;                 // A row (M)
    const int nc   = lane & 15;                 // B col (N)

    const float* Ap = Wx + (size_t)e * (NC * NC) + (size_t)(m0 + mr) * NC;
    const float* Bp = x  + (size_t)b * NC * HW + (size_t)n0 + nc;

    v8f acc = {};
    #pragma unroll
    for (int k0 = 0; k0 < NC; k0 += 4) {
        const int ka = k0 + 2 * half;
        v2f av, bv;
        av.x = Ap[ka];
        av.y = Ap[ka + 1];
        bv.x = Bp[(size_t)ka * HW];
        bv.y = Bp[(size_t)(ka + 1) * HW];
        acc = __builtin_amdgcn_wmma_f32_16x16x4_f32(
                  false, av, false, bv, (short)0, acc, false, false);
    }

    float* op = out + ((size_t)b * NC + m0 + 8 * half) * HW + n0 + nc;
    #pragma unroll
    for (int i = 0; i < 8; ++i)
        op[(size_t)i * HW] = acc[i] * sc;
}

// ---------------------------------------------------------------------------
extern "C" void kernel_launch(void* const* d_in, const int* in_sizes, int n_in,
                              void* d_out, int out_size, void* d_ws, size_t ws_size,
                              hipStream_t stream)
{
    const float* x        = (const float*)d_in[0];   // [32,64,64,64]
    const float* gate_w   = (const float*)d_in[1];   // [8,64,3,3]
    const float* gate_b   = (const float*)d_in[2];   // [8]
    const float* expert_w = (const float*)d_in[3];   // [8,64,64]

    float* out_main = (float*)d_out;                               // [32,64,64,64]
    float* out_ew   = (float*)d_out + (size_t)NB * NC * HW;        // [32,8]

    // workspace layout
    float* S     = (float*)d_ws;                                   // 2048*9 floats
    float* scale = (float*)((char*)d_ws + (size_t)NB * NC * 9 * 4);// 32 floats
    int*   eidx  = (int*)(scale + NB);                             // 32 ints

    k_window_sums<<<NB * NC, 64, 0, stream>>>(x, S);
    k_gate_select<<<NB, 64, 0, stream>>>(S, gate_w, gate_b, out_ew, scale, eidx);
    k_expert_gemm<<<4096, 256, 0, stream>>>(x, expert_w, scale, eidx, out_main);
}